// DigitCaps_39943195853313
// MI455X (gfx1250) — compile-verified
//
#include <hip/hip_runtime.h>
#include <hip/hip_bf16.h>
#include <math.h>

// Problem constants (from reference)
#define B_   256
#define R_   1152
#define C_   10
#define IN_  8
#define OUT_ 16
#define CO_  (C_ * OUT_)          // 160
#define RC_  (R_ * C_)            // 11520

typedef __attribute__((ext_vector_type(2))) float v2f;
typedef __attribute__((ext_vector_type(8))) float v8f;

// ---------------------------------------------------------------------------
// Kernel 1: u_hat[b,r,c,o] = sum_i W[r,c,o,i] * x[b,r,i]
// Per route r: GEMM  (B x IN) @ (IN x CO)  ->  (B x CO), K=8.
// One wave (32 threads) per (r, 16-row batch tile); V_WMMA_F32_16X16X4_F32,
// two K=4 steps, 10 N-tiles of 16.
//
// A fragment (16x4 f32): lanes 0-15 -> M=lane, VGPR0=K0, VGPR1=K1;
//                        lanes 16-31 -> M=lane-16, VGPR0=K2, VGPR1=K3.
// B fragment (4x16 f32): lanes 0-15 -> N=lane, VGPR0=K0, VGPR1=K1;
//                        lanes 16-31 -> N=lane-16, VGPR0=K2, VGPR1=K3.
// D (16x16 f32, 8 VGPRs): VGPR v -> M=v (lanes 0-15) / M=v+8 (lanes 16-31),
//                         N = lane & 15.
// ---------------------------------------------------------------------------
__global__ __launch_bounds__(32) void uhat_kernel(const float* __restrict__ x,
                                                  const float* __restrict__ W,
                                                  float* __restrict__ uhat) {
    const int r     = blockIdx.x;      // 0..R-1
    const int bt    = blockIdx.y;      // 0..B/16-1
    const int lane  = threadIdx.x;     // 0..31
    const int nlow  = lane & 15;
    const int khalf = lane >> 4;       // 0 or 1
    const int kb    = khalf * 2;       // K base within a K=4 step

    // A fragments: x[b, r, i] ; frag1 covers i=0..3, frag2 covers i=4..7
    const int mrow = bt * 16 + nlow;
    const float* xp = x + (size_t)mrow * (R_ * IN_) + (size_t)r * IN_ + kb;
    v2f a1, a2;
    a1.x = xp[0]; a1.y = xp[1];
    a2.x = xp[4]; a2.y = xp[5];

    // B fragments: W[r, c, o, i] with n = c*16+o; column n = 8 contiguous i's
    const float* wp = W + (size_t)r * (C_ * OUT_ * IN_) + (size_t)nlow * IN_ + kb;
    const int mbase = bt * 16 + khalf * 8;

    for (int nt = 0; nt < C_; ++nt) {   // 10 N-tiles of 16 (one capsule each)
        const float* wq = wp + nt * (OUT_ * IN_);   // +128 floats per tile
        v2f b1, b2;
        b1.x = wq[0]; b1.y = wq[1];
        b2.x = wq[4]; b2.y = wq[5];

        v8f acc = {};
        acc = __builtin_amdgcn_wmma_f32_16x16x4_f32(false, a1, false, b1,
                                                    (short)0, acc, false, false);
        acc = __builtin_amdgcn_wmma_f32_16x16x4_f32(false, a2, false, b2,
                                                    (short)0, acc, false, false);

        const int ncol = nt * 16 + nlow;            // = c*16 + o
#pragma unroll
        for (int v = 0; v < 8; ++v) {
            const size_t brow = (size_t)(mbase + v) * R_ + r;
            uhat[brow * CO_ + ncol] = acc[v];
        }
    }
}

// ---------------------------------------------------------------------------
// Kernel 2: c[r,c] = softmax over r of b_logits[r,c]   (one block per capsule)
// ---------------------------------------------------------------------------
__global__ __launch_bounds__(256) void softmax_kernel(const float* __restrict__ blog,
                                                      float* __restrict__ cvals) {
    const int c = blockIdx.x;          // 0..C-1
    const int t = threadIdx.x;
    __shared__ float red[256];

    float m = -3.4e38f;
    for (int r = t; r < R_; r += 256) m = fmaxf(m, blog[r * C_ + c]);
    red[t] = m;
    __syncthreads();
    for (int s = 128; s > 0; s >>= 1) {
        if (t < s) red[t] = fmaxf(red[t], red[t + s]);
        __syncthreads();
    }
    m = red[0];
    __syncthreads();

    float sum = 0.0f;
    for (int r = t; r < R_; r += 256) sum += expf(blog[r * C_ + c] - m);
    red[t] = sum;
    __syncthreads();
    for (int s = 128; s > 0; s >>= 1) {
        if (t < s) red[t] += red[t + s];
        __syncthreads();
    }
    const float inv = 1.0f / red[0];

    for (int r = t; r < R_; r += 256)
        cvals[r * C_ + c] = expf(blog[r * C_ + c] - m) * inv;
}

// ---------------------------------------------------------------------------
// Kernel 3: s[b,c,o] = sum_r c[r,c]*u_hat[b,r,c,o] ; v = squash(s) (elementwise)
// One block per batch element b; 160 threads = (c,o). Coalesced 640B rows.
// ---------------------------------------------------------------------------
__global__ __launch_bounds__(160) void sv_kernel(const float* __restrict__ uhat,
                                                 const float* __restrict__ cvals,
                                                 float* __restrict__ vout) {
    const int b = blockIdx.x;
    const int t = threadIdx.x;         // c*16 + o
    const int c = t >> 4;

    const float* up = uhat + (size_t)b * (R_ * CO_) + t;
    float acc = 0.0f;
    for (int r = 0; r < R_; ++r)
        acc = fmaf(cvals[r * C_ + c], up[(size_t)r * CO_], acc);

    // reference squash is elementwise: v = s^2 / ((1+s^2)*sqrt(s^2)) * s
    const float s2 = acc * acc;
    const float v  = s2 / ((1.0f + s2) * sqrtf(s2)) * acc;
    vout[b * CO_ + t] = v;
}

// ---------------------------------------------------------------------------
// Kernel 4: b_logits[r,c] += (1/B) * sum_b sum_o u_hat[b,r,c,o]*v[b,c,o]
// One block per route r; 160 threads = (c,o); serial b-loop (deterministic),
// then a 16-lane shfl butterfly reduces over o.
// ---------------------------------------------------------------------------
__global__ __launch_bounds__(160) void agree_kernel(const float* __restrict__ uhat,
                                                    const float* __restrict__ vbuf,
                                                    float* __restrict__ blog) {
    const int r = blockIdx.x;
    const int t = threadIdx.x;         // c*16 + o

    const float* up = uhat + (size_t)r * CO_ + t;
    float acc = 0.0f;
    for (int b = 0; b < B_; ++b)
        acc = fmaf(up[(size_t)b * (R_ * CO_)], vbuf[b * CO_ + t], acc);

    // sum over o within each 16-lane group (lanes are consecutive o)
    acc += __shfl_xor(acc, 1, 32);
    acc += __shfl_xor(acc, 2, 32);
    acc += __shfl_xor(acc, 4, 32);
    acc += __shfl_xor(acc, 8, 32);

    if ((t & 15) == 0)
        blog[r * C_ + (t >> 4)] += acc * (1.0f / (float)B_);
}

__global__ void zero_kernel(float* __restrict__ p, int n) {
    const int i = blockIdx.x * blockDim.x + threadIdx.x;
    if (i < n) p[i] = 0.0f;
}

// ---------------------------------------------------------------------------
extern "C" void kernel_launch(void* const* d_in, const int* in_sizes, int n_in,
                              void* d_out, int out_size, void* d_ws, size_t ws_size,
                              hipStream_t stream) {
    const float* x = (const float*)d_in[0];   // (B, R, IN)
    const float* W = (const float*)d_in[1];   // (R, C, OUT, IN)
    float* vout = (float*)d_out;              // (B, C, OUT, 1)

    float* uhat  = (float*)d_ws;                              // B*R*C*OUT
    float* blog  = uhat + (size_t)B_ * R_ * CO_;              // R*C logits
    float* cvals = blog + RC_;                                // R*C softmax
    float* vbuf  = cvals + RC_;                               // B*C*OUT

    zero_kernel<<<(RC_ + 255) / 256, 256, 0, stream>>>(blog, RC_);

    uhat_kernel<<<dim3(R_, B_ / 16), 32, 0, stream>>>(x, W, uhat);

    for (int it = 0; it < 3; ++it) {
        softmax_kernel<<<C_, 256, 0, stream>>>(blog, cvals);
        sv_kernel<<<B_, 160, 0, stream>>>(uhat, cvals, (it == 2) ? vout : vbuf);
        if (it < 2)
            agree_kernel<<<R_, 160, 0, stream>>>(uhat, vbuf, blog);
    }
}